// RotatedPSRoIAlign_84765474554608
// MI455X (gfx1250) — compile-verified
//
#include <hip/hip_runtime.h>
#include <hip/hip_bf16.h>

typedef __attribute__((ext_vector_type(2))) float v2f;
typedef __attribute__((ext_vector_type(8))) float v8f;

#define FM_B 8
#define FM_C 256
#define FM_H 200
#define FM_W 200
#define ROI_N_FIELDS 6
#define OUT_HW 49   // 7x7

// ---------------------------------------------------------------------------
// Kernel 1: NCHW -> NHWC transpose of the feature map into workspace.
// grid: (ceil(W/32), C/32, B*H), block: (32, 8). LDS-tiled, padded to kill
// bank conflicts. Reads coalesced along W, writes coalesced along C.
// ---------------------------------------------------------------------------
__global__ __launch_bounds__(256)
void nchw_to_nhwc_kernel(const float* __restrict__ in, float* __restrict__ out) {
  __shared__ float tile[32][33];
  const int by = blockIdx.z;          // b*H + y
  const int b  = by / FM_H;
  const int y  = by % FM_H;
  const int c0 = blockIdx.y * 32;
  const int x0 = blockIdx.x * 32;

#pragma unroll
  for (int i = 0; i < 4; ++i) {
    const int c = c0 + threadIdx.y + i * 8;
    const int x = x0 + threadIdx.x;
    if (x < FM_W)
      tile[threadIdx.y + i * 8][threadIdx.x] =
          in[((b * FM_C + c) * FM_H + y) * FM_W + x];
  }
  __syncthreads();
#pragma unroll
  for (int i = 0; i < 4; ++i) {
    const int x = x0 + threadIdx.y + i * 8;
    const int c = c0 + threadIdx.x;
    if (x < FM_W)
      out[(by * FM_W + x) * FM_C + c] = tile[threadIdx.x][threadIdx.y + i * 8];
  }
}

// ---------------------------------------------------------------------------
// Kernel 2: WMMA rotated bilinear sampling from the NHWC buffer.
// One wave (32 threads) per (roi, group-of-16-points). 49 points -> 4 groups
// (group 3 runs only k-step 0 since only point 48 exists).
// Out(16 pts x 16 ch) = W(16x64) x V(64x16), K split into 16 chunks of 4
// (one point's 4 bilinear corners per chunk) via V_WMMA_F32_16X16X4_F32.
//
// Per k-step: all 16 channel-tiles' B operands are loaded into a live
// register array first (one 32-load clause, one wait), then 16 WMMAs issue
// back-to-back; the next step's corner rows are prefetched meanwhile.
// ---------------------------------------------------------------------------
__global__ __launch_bounds__(32)
void rroi_align_wmma_kernel(const float* __restrict__ T,
                            const float* __restrict__ rois,
                            float* __restrict__ out, int N) {
  const int wid = blockIdx.x;
  const int r = wid >> 2;
  const int g = wid & 3;
  if (r >= N) return;

  const int l  = threadIdx.x;
  const int m  = l & 15;            // row within tile (both halves of wave)
  const bool hi = l >= 16;

  // ROI parameters (uniform across wave)
  const int   b   = (int)rois[r * ROI_N_FIELDS + 0];
  const float xc  = rois[r * ROI_N_FIELDS + 1];
  const float yc  = rois[r * ROI_N_FIELDS + 2];
  const float ang = rois[r * ROI_N_FIELDS + 5];
  const float cv = cosf(ang);
  const float sv = sinf(ang);

  // Per-lane sample point: p = g*16 + m (lanes 16-31 mirror lanes 0-15)
  const int  p      = g * 16 + m;
  const bool pvalid = p < OUT_HW;
  const int  ph = p / 7;
  const int  pw = p % 7;
  const float gy = -1.0f + (float)ph * (2.0f / 6.0f);
  const float gx = -1.0f + (float)pw * (2.0f / 6.0f);
  const float xn = cv * gx - sv * gy + xc;
  const float yn = sv * gx + cv * gy + yc;
  const float ix = (xn + 1.0f) * 0.5f * (float)(FM_W - 1);
  const float iy = (yn + 1.0f) * 0.5f * (float)(FM_H - 1);
  const float x0f = floorf(ix), y0f = floorf(iy);
  const float wx1 = ix - x0f, wx0 = 1.0f - wx1;
  const float wy1 = iy - y0f, wy0 = 1.0f - wy1;
  const int x0 = (int)x0f, y0 = (int)y0f;
  const int x1 = x0 + 1,  y1 = y0 + 1;
  const bool vx0 = (x0 >= 0) && (x0 <= FM_W - 1);
  const bool vx1 = (x1 >= 0) && (x1 <= FM_W - 1);
  const bool vy0 = (y0 >= 0) && (y0 <= FM_H - 1);
  const bool vy1 = (y1 >= 0) && (y1 <= FM_H - 1);
  const int xs0 = min(max(x0, 0), FM_W - 1);
  const int xs1 = min(max(x1, 0), FM_W - 1);
  const int ys0 = min(max(y0, 0), FM_H - 1);
  const int ys1 = min(max(y1, 0), FM_H - 1);

  const float pv = pvalid ? 1.0f : 0.0f;
  // Corner order k: (y0,x0), (y0,x1), (y1,x0), (y1,x1)
  const float w0 = wy0 * wx0 * ((vy0 && vx0) ? pv : 0.0f);
  const float w1 = wy0 * wx1 * ((vy0 && vx1) ? pv : 0.0f);
  const float w2 = wy1 * wx0 * ((vy1 && vx0) ? pv : 0.0f);
  const float w3 = wy1 * wx1 * ((vy1 && vx1) ? pv : 0.0f);
  // Element offsets into NHWC buffer (channels contiguous)
  const int base_bh = b * FM_H;
  const int off0 = ((base_bh + ys0) * FM_W + xs0) * FM_C;
  const int off1 = ((base_bh + ys0) * FM_W + xs1) * FM_C;
  const int off2 = ((base_bh + ys1) * FM_W + xs0) * FM_C;
  const int off3 = ((base_bh + ys1) * FM_W + xs1) * FM_C;

  // Lane-selected operands matching WMMA K-row placement:
  // a.x/b.x -> K0 (lanes 0-15) / K2 (lanes 16-31); a.y/b.y -> K1 / K3.
  const int   offA = hi ? off2 : off0;
  const int   offB = hi ? off3 : off1;
  const float wA   = hi ? w2 : w0;
  const float wB   = hi ? w3 : w1;

  // Group 3 holds only point 48 -> only k-step 0 contributes.
  const int smax = (g == 3) ? 1 : 16;

  v8f acc[16] = {};   // 16 channel tiles of 16x16 f32 accumulators

  // Prefetch step-0 corner rows (1KB each, lanes spread at 64B stride).
  {
    const int o0A = __shfl(offA, 0 | (l & 16), 32);
    const int o0B = __shfl(offB, 0 | (l & 16), 32);
    __builtin_prefetch(T + o0A + m * 16, 0, 0);
    __builtin_prefetch(T + o0B + m * 16, 0, 0);
  }

  for (int s = 0; s < smax; ++s) {
    // Broadcast point s's corner offsets (low lanes pull from lane s,
    // high lanes from lane s+16 -> ds_bpermute).
    const int src = s | (l & 16);
    const int oA = __shfl(offA, src, 32);
    const int oB = __shfl(offB, src, 32);
    const float* pA = T + (oA + m);
    const float* pB = T + (oB + m);

    // Phase 1: fetch all 16 channel tiles' B operands into live registers
    // (single wide load clause, single wait).
    v2f bm[16];
#pragma unroll
    for (int ct = 0; ct < 16; ++ct) {
      bm[ct].x = pA[ct * 16];
      bm[ct].y = pB[ct * 16];
    }

    // Prefetch next step's corner rows while this step's WMMAs run.
    if (s + 1 < smax) {
      const int onA = __shfl(offA, (s + 1) | (l & 16), 32);
      const int onB = __shfl(offB, (s + 1) | (l & 16), 32);
      __builtin_prefetch(T + onA + m * 16, 0, 0);
      __builtin_prefetch(T + onB + m * 16, 0, 0);
    }

    // A operand: only row m == s carries nonzero weights.
    v2f a;
    a.x = (m == s) ? wA : 0.0f;
    a.y = (m == s) ? wB : 0.0f;

    // Phase 2: 16 back-to-back WMMAs.
#pragma unroll
    for (int ct = 0; ct < 16; ++ct) {
      acc[ct] = __builtin_amdgcn_wmma_f32_16x16x4_f32(
          /*neg_a=*/false, a, /*neg_b=*/false, bm[ct],
          /*c_mod=*/(short)0, acc[ct], /*reuse_a=*/false, /*reuse_b=*/false);
    }
  }

  // Store: D tile lane l, VGPR v -> point row M = v + (hi?8:0), channel
  // N = l&15. Output layout (N, C, 7, 7).
  const int rowoff = hi ? 8 : 0;
#pragma unroll
  for (int ct = 0; ct < 16; ++ct) {
    const int c = ct * 16 + m;
    float* orow = out + ((size_t)r * FM_C + c) * OUT_HW;
#pragma unroll
    for (int v = 0; v < 8; ++v) {
      const int pp = g * 16 + v + rowoff;
      if (pp < OUT_HW) orow[pp] = acc[ct][v];
    }
  }
}

// ---------------------------------------------------------------------------
// Fallback: direct gather from NCHW (used only if workspace is too small).
// One block of 256 threads per (roi, point); thread = channel.
// ---------------------------------------------------------------------------
__global__ __launch_bounds__(256)
void rroi_align_direct_kernel(const float* __restrict__ fm,
                              const float* __restrict__ rois,
                              float* __restrict__ out, int N) {
  const int rp = blockIdx.x;
  const int r = rp / OUT_HW;
  const int p = rp % OUT_HW;
  if (r >= N) return;
  const int c = threadIdx.x;

  const int   b   = (int)rois[r * ROI_N_FIELDS + 0];
  const float xc  = rois[r * ROI_N_FIELDS + 1];
  const float yc  = rois[r * ROI_N_FIELDS + 2];
  const float ang = rois[r * ROI_N_FIELDS + 5];
  const float cv = cosf(ang), sv = sinf(ang);

  const int ph = p / 7, pw = p % 7;
  const float gy = -1.0f + (float)ph * (2.0f / 6.0f);
  const float gx = -1.0f + (float)pw * (2.0f / 6.0f);
  const float xn = cv * gx - sv * gy + xc;
  const float yn = sv * gx + cv * gy + yc;
  const float ix = (xn + 1.0f) * 0.5f * (float)(FM_W - 1);
  const float iy = (yn + 1.0f) * 0.5f * (float)(FM_H - 1);
  const float x0f = floorf(ix), y0f = floorf(iy);
  const float wx1 = ix - x0f, wx0 = 1.0f - wx1;
  const float wy1 = iy - y0f, wy0 = 1.0f - wy1;
  const int x0 = (int)x0f, y0 = (int)y0f, x1 = x0 + 1, y1 = y0 + 1;
  const bool vx0 = (x0 >= 0) && (x0 < FM_W), vx1 = (x1 >= 0) && (x1 < FM_W);
  const bool vy0 = (y0 >= 0) && (y0 < FM_H), vy1 = (y1 >= 0) && (y1 < FM_H);
  const int xs0 = min(max(x0, 0), FM_W - 1), xs1 = min(max(x1, 0), FM_W - 1);
  const int ys0 = min(max(y0, 0), FM_H - 1), ys1 = min(max(y1, 0), FM_H - 1);
  const float w00 = wy0 * wx0 * ((vy0 && vx0) ? 1.0f : 0.0f);
  const float w01 = wy0 * wx1 * ((vy0 && vx1) ? 1.0f : 0.0f);
  const float w10 = wy1 * wx0 * ((vy1 && vx0) ? 1.0f : 0.0f);
  const float w11 = wy1 * wx1 * ((vy1 && vx1) ? 1.0f : 0.0f);

  const float* plane = fm + ((size_t)b * FM_C + c) * (FM_H * FM_W);
  float acc = w00 * plane[ys0 * FM_W + xs0] + w01 * plane[ys0 * FM_W + xs1] +
              w10 * plane[ys1 * FM_W + xs0] + w11 * plane[ys1 * FM_W + xs1];
  out[((size_t)r * FM_C + c) * OUT_HW + p] = acc;
}

// ---------------------------------------------------------------------------
extern "C" void kernel_launch(void* const* d_in, const int* in_sizes, int n_in,
                              void* d_out, int out_size, void* d_ws,
                              size_t ws_size, hipStream_t stream) {
  const float* fm   = (const float*)d_in[0];
  const float* rois = (const float*)d_in[1];
  float* out = (float*)d_out;
  const int N = in_sizes[1] / ROI_N_FIELDS;

  const size_t nhwc_bytes =
      (size_t)FM_B * FM_C * FM_H * FM_W * sizeof(float);

  if (ws_size >= nhwc_bytes) {
    float* T = (float*)d_ws;
    dim3 tb(32, 8);
    dim3 tg((FM_W + 31) / 32, FM_C / 32, FM_B * FM_H);
    nchw_to_nhwc_kernel<<<tg, tb, 0, stream>>>(fm, T);
    rroi_align_wmma_kernel<<<N * 4, 32, 0, stream>>>(T, rois, out, N);
  } else {
    rroi_align_direct_kernel<<<N * OUT_HW, 256, 0, stream>>>(fm, rois, out, N);
  }
}